// SwitchGate_90529320665228
// MI455X (gfx1250) — compile-verified
//
#include <hip/hip_runtime.h>

typedef __attribute__((ext_vector_type(2))) float v2f;
typedef __attribute__((ext_vector_type(4))) float v4f;
typedef __attribute__((ext_vector_type(8))) float v8f;

namespace {
constexpr int   kE            = 8;       // experts
constexpr int   kD            = 2048;    // feature dim
constexpr int   kB            = 65536;   // tokens
constexpr int   kThreads      = 256;     // 8 waves
constexpr int   kRowsPerWave  = 32;      // two 16-row WMMA tiles per wave
constexpr int   kRowsPerBlock = 256;     // 8 waves * 32 rows
constexpr float kEps          = 1e-6f;
constexpr float kCapacity     = 65536.0f; // int(1.0 * B)
}

// Pass 1: WMMA gate GEMM + bias + domain mask + softmax + top-1 one-hot;
// writes masked probabilities to `out` and accumulates per-expert column sums.
__global__ __launch_bounds__(kThreads)
void switch_gate_wmma(const float* __restrict__ x,      // [B, D]
                      const int*   __restrict__ label,  // [B]
                      const float* __restrict__ wg,     // [D, E]
                      const float* __restrict__ bg,     // [E]
                      float*       __restrict__ out,    // [B, E]
                      float*       __restrict__ denom)  // [E] (pre-zeroed)
{
    __shared__ float lds[kD * kE];  // 64 KB: W during GEMM, score exchange after

    const int tid  = threadIdx.x;
    const int lane = tid & 31;       // wave32
    const int wv   = tid >> 5;       // 8 waves per block
    const int hi   = lane >> 4;      // lane half selects K pair (ISA 7.12.2)
    const int col  = lane & 15;
    const int wcol = col & (kE - 1); // cols 8-15 replicate 0-7 (LDS broadcast, no
                                     // divergent branch; duplicate C cols discarded)

    // Stage w_gate (2048x8 fp32 = 64 KB) into LDS, coalesced b128 loads.
    {
        const v4f* s = (const v4f*)wg;
        v4f*       d = (v4f*)lds;
        #pragma unroll
        for (int i = 0; i < (kD * kE / 4) / kThreads; ++i)
            d[tid + i * kThreads] = s[tid + i * kThreads];
    }
    __syncthreads();

    // Each wave: two 16-row tiles sharing the same B fragment (2 WMMA per LDS read).
    const int    rowBase = blockIdx.x * kRowsPerBlock + wv * kRowsPerWave;
    const float* xp0 = x + (size_t)(rowBase + col) * kD + 2 * hi;
    const float* xp1 = xp0 + (size_t)16 * kD;

    v8f c0 = {0.f, 0.f, 0.f, 0.f, 0.f, 0.f, 0.f, 0.f};
    v8f c1 = {0.f, 0.f, 0.f, 0.f, 0.f, 0.f, 0.f, 0.f};

    #pragma unroll 4
    for (int k = 0; k < kD; k += 4) {
        // A fragments (16x4 f32): lanes 0-15 hold K=k..k+1, lanes 16-31 K=k+2..k+3
        v2f a0 = *(const v2f*)(xp0 + k);
        v2f a1 = *(const v2f*)(xp1 + k);
        // B fragment (4x16 f32): lane holds column; VGPR0<->K=k+2*hi, VGPR1<->K+1
        const float* wp = lds + (k + 2 * hi) * kE + wcol;
        v2f b;
        b[0] = wp[0];
        b[1] = wp[kE];
        c0 = __builtin_amdgcn_wmma_f32_16x16x4_f32(
            false, a0, false, b, (short)0, c0, false, false);
        c1 = __builtin_amdgcn_wmma_f32_16x16x4_f32(
            false, a1, false, b, (short)0, c1, false, false);
    }
    __syncthreads();  // all waves finished reading W -> reuse LDS

    float* sc   = lds;                        // scores [256][8]
    float* pacc = lds + kRowsPerBlock * kE;   // per-expert partials [8]
    if (tid < kE) pacc[tid] = 0.f;

    // Scatter C tiles: lane holds column `col`; VGPR v holds row v + 8*hi.
    if (col < kE) {
        const int r0 = wv * kRowsPerWave + hi * 8;
        #pragma unroll
        for (int v = 0; v < 8; ++v)
            sc[(r0 + v) * kE + col] = c0[v];
        #pragma unroll
        for (int v = 0; v < 8; ++v)
            sc[(r0 + 16 + v) * kE + col] = c1[v];
    }
    __syncthreads();

    // One thread per row (256 threads == 256 rows): mask, softmax, top-1.
    {
        const int grow = blockIdx.x * kRowsPerBlock + tid;
        const int off  = label[grow] * (kE / 2);  // domain window: [0,4) or [4,8)

        float r0 = sc[tid * kE + off + 0] + bg[off + 0];
        float r1 = sc[tid * kE + off + 1] + bg[off + 1];
        float r2 = sc[tid * kE + off + 2] + bg[off + 2];
        float r3 = sc[tid * kE + off + 3] + bg[off + 3];

        float mx = r0; int am = 0;                 // strict '>' => first-max, like jnp.argmax
        if (r1 > mx) { mx = r1; am = 1; }
        if (r2 > mx) { mx = r2; am = 2; }
        if (r3 > mx) { mx = r3; am = 3; }

        const float s = expf(r0 - mx) + expf(r1 - mx) + expf(r2 - mx) + expf(r3 - mx);
        const float p = 1.0f / s;                  // softmax value at the argmax
        const int  t1 = off + am;

        v4f lo, hi4;
        lo.x  = (t1 == 0) ? p : 0.f;  lo.y  = (t1 == 1) ? p : 0.f;
        lo.z  = (t1 == 2) ? p : 0.f;  lo.w  = (t1 == 3) ? p : 0.f;
        hi4.x = (t1 == 4) ? p : 0.f;  hi4.y = (t1 == 5) ? p : 0.f;
        hi4.z = (t1 == 6) ? p : 0.f;  hi4.w = (t1 == 7) ? p : 0.f;

        v4f* orow = (v4f*)(out + (size_t)grow * kE);
        orow[0] = lo;
        orow[1] = hi4;

        atomicAdd(&pacc[t1], p);   // ds_add_f32
    }
    __syncthreads();

    if (tid < kE) atomicAdd(&denom[tid], pacc[tid]);  // one global atomic per expert/block
}

// Pass 2: out[b][e] *= capacity / (denom[e] + eps), in place. 2 MB r+w -> negligible.
__global__ __launch_bounds__(kThreads)
void switch_gate_finalize(float* __restrict__ out, const float* __restrict__ denom)
{
    const float s0 = kCapacity / (denom[0] + kEps);
    const float s1 = kCapacity / (denom[1] + kEps);
    const float s2 = kCapacity / (denom[2] + kEps);
    const float s3 = kCapacity / (denom[3] + kEps);
    const float s4 = kCapacity / (denom[4] + kEps);
    const float s5 = kCapacity / (denom[5] + kEps);
    const float s6 = kCapacity / (denom[6] + kEps);
    const float s7 = kCapacity / (denom[7] + kEps);

    v4f* o = (v4f*)out;
    const int stride = gridDim.x * blockDim.x;
    for (int r = blockIdx.x * blockDim.x + threadIdx.x; r < kB; r += stride) {
        v4f a = o[2 * r + 0];
        v4f b = o[2 * r + 1];
        a.x *= s0; a.y *= s1; a.z *= s2; a.w *= s3;
        b.x *= s4; b.y *= s5; b.z *= s6; b.w *= s7;
        o[2 * r + 0] = a;
        o[2 * r + 1] = b;
    }
}

extern "C" void kernel_launch(void* const* d_in, const int* in_sizes, int n_in,
                              void* d_out, int out_size, void* d_ws, size_t ws_size,
                              hipStream_t stream)
{
    const float* x     = (const float*)d_in[0];
    const int*   label = (const int*)  d_in[1];
    const float* wg    = (const float*)d_in[2];
    const float* bg    = (const float*)d_in[3];
    float*       out   = (float*)d_out;
    float*       denom = (float*)d_ws;

    hipMemsetAsync(denom, 0, kE * sizeof(float), stream);

    switch_gate_wmma<<<kB / kRowsPerBlock, kThreads, 0, stream>>>(
        x, label, wg, bg, out, denom);

    switch_gate_finalize<<<256, kThreads, 0, stream>>>(out, denom);
}